// KnowledgeEmbeddings_80839874445880
// MI455X (gfx1250) — compile-verified
//
#include <hip/hip_runtime.h>

#define WORD_LEN 256
#define KN_LEN   64
#define SEQ      320
#define HID      768
#define VOCAB    30522
#define D_ENT    100
#define B_SZ     64
#define LN_EPS   1e-12f

typedef __attribute__((ext_vector_type(2))) float v2f;
typedef __attribute__((ext_vector_type(8))) float v8f;

// ---------------------------------------------------------------------------
// Kernel 1: per-row stable compaction of word / entity tokens.
// One block per batch row; lane 0 does the (trivial) 320-element serial scan
// in LDS, then all 320 threads write the compacted arrays out coalesced.
// ---------------------------------------------------------------------------
__global__ void __launch_bounds__(SEQ) compact_kernel(
    const int* __restrict__ ids, const int* __restrict__ tts,
    int* __restrict__ wid, int* __restrict__ wtt, int* __restrict__ wpos,
    int* __restrict__ kid, int* __restrict__ ktt, int* __restrict__ kpos,
    int* __restrict__ kval)
{
    __shared__ int sid[SEQ], stt[SEQ];
    __shared__ int swid[WORD_LEN], swtt[WORD_LEN], swpos[WORD_LEN];
    __shared__ int skid[KN_LEN], sktt[KN_LEN], skpos[KN_LEN], skval[KN_LEN];
    const int b = blockIdx.x;
    const int t = threadIdx.x;
    sid[t] = ids[b * SEQ + t];
    stt[t] = tts[b * SEQ + t];
    __syncthreads();
    if (t == 0) {
        int wc = 0, kc = 0;
        for (int i = 0; i < SEQ; ++i) {
            int v = sid[i];
            if (v > 0 && v < VOCAB) {
                if (wc < WORD_LEN) { swid[wc] = v; swtt[wc] = stt[i]; swpos[wc] = i; ++wc; }
            } else if (v >= VOCAB) {
                if (kc < KN_LEN) { skid[kc] = v - VOCAB; sktt[kc] = stt[i]; skpos[kc] = i; skval[kc] = 1; ++kc; }
            }
        }
        for (int i = wc; i < WORD_LEN; ++i) { swid[i] = 0; swtt[i] = 1; swpos[i] = i; }
        for (int i = kc; i < KN_LEN;  ++i) { skid[i] = 0; sktt[i] = 0; skpos[i] = 0; skval[i] = 0; }
    }
    __syncthreads();
    if (t < WORD_LEN) {
        wid[b * WORD_LEN + t]  = swid[t];
        wtt[b * WORD_LEN + t]  = swtt[t];
        wpos[b * WORD_LEN + t] = swpos[t];
    }
    if (t < KN_LEN) {
        kid[b * KN_LEN + t]  = skid[t];
        ktt[b * KN_LEN + t]  = sktt[t];
        kpos[b * KN_LEN + t] = skpos[t];
        kval[b * KN_LEN + t] = skval[t];
    }
}

// ---------------------------------------------------------------------------
// Kernel 2: word-side gather-sum + LayerNorm. One wave32 per token row;
// 24 f32 elements per lane, shfl_xor reduction over the 32-lane wave.
// Pure bandwidth: ~100 MB total at 23.3 TB/s.
// ---------------------------------------------------------------------------
__global__ void __launch_bounds__(256) word_kernel(
    const float* __restrict__ word_emb, const float* __restrict__ pos_emb,
    const float* __restrict__ tt_emb,   const float* __restrict__ gamma,
    const float* __restrict__ beta,
    const int* __restrict__ wid, const int* __restrict__ wtt,
    const int* __restrict__ wpos, float* __restrict__ out)
{
    const int lane = threadIdx.x & 31;
    const int wave = threadIdx.x >> 5;
    const int row  = blockIdx.x * 8 + wave;     // 0 .. B*WORD_LEN-1
    const int b    = row / WORD_LEN;
    const int p    = row % WORD_LEN;
    const float* we = word_emb + (size_t)wid[row] * HID;
    const float* pe = pos_emb  + (size_t)wpos[row] * HID;
    const float* te = tt_emb   + (size_t)wtt[row]  * HID;

    float x[HID / 32];
    float s = 0.f, sq = 0.f;
#pragma unroll
    for (int t = 0; t < HID / 32; ++t) {
        const int j = lane + t * 32;
        const float v = we[j] + te[j] + pe[j];
        x[t] = v; s += v; sq += v * v;
    }
#pragma unroll
    for (int o = 16; o > 0; o >>= 1) {
        s  += __shfl_xor(s,  o, 32);
        sq += __shfl_xor(sq, o, 32);
    }
    const float mean = s * (1.f / HID);
    const float var  = sq * (1.f / HID) - mean * mean;
    const float rstd = rsqrtf(var + LN_EPS);
    float* op = out + ((size_t)b * SEQ + p) * HID;
#pragma unroll
    for (int t = 0; t < HID / 32; ++t) {
        const int j = lane + t * 32;
        op[j] = gamma[j] * (x[t] - mean) * rstd + beta[j];
    }
}

// ---------------------------------------------------------------------------
// Kernel 3: knowledge side, fused GEMM (V_WMMA_F32_16X16X4_F32) + bias/emb
// + mask + LayerNorm. Block = 16 waves = one 16-row M-tile; each wave does
// 3 of the 48 N-tiles (25 WMMA K-steps each, K=100), parks the 16x16 f32
// accumulator in a 16x768 LDS tile, then wave w LayerNorms row w.
// A-fragment: lanes 0-15 K={0,1}, lanes 16-31 K={2,3}; B mirrored by N.
// C/D: VGPR v -> row v + 8*(lane>=16), col lane&15.
// ---------------------------------------------------------------------------
__global__ void __launch_bounds__(512) kn_kernel(
    const float* __restrict__ entityVec, const float* __restrict__ ke_w,
    const float* __restrict__ ke_b,      const float* __restrict__ tt_emb,
    const float* __restrict__ pos_emb,   const float* __restrict__ gamma,
    const float* __restrict__ beta,
    const int* __restrict__ kid,  const int* __restrict__ ktt,
    const int* __restrict__ kpos, const int* __restrict__ kval,
    float* __restrict__ out)
{
    __shared__ float cTile[16][HID + 8];
    const int lane = threadIdx.x & 31;
    const int wave = threadIdx.x >> 5;      // 0..15
    const int tile = blockIdx.x;            // M-tile: 16 entity-token rows
    const int half = lane >> 4;             // 0/1
    const int lm   = lane & 15;

    // ---- phase 1: WMMA GEMM  ev[16,100] x ke_w^T[100,16] per tile ----
    const int   rA    = tile * 16 + lm;                       // A row (M)
    const float amask = (float)kval[rA];                      // zero padded rows
    const float* aRow = entityVec + (size_t)kid[rA] * D_ENT;
    const int   koff  = half * 2;                             // K sub-slab

    for (int nt = wave * 3; nt < wave * 3 + 3; ++nt) {
        const int n0 = nt * 16;
        const float* bRow = ke_w + (size_t)(n0 + lm) * D_ENT; // B col (N)
        v8f c = {};
#pragma unroll
        for (int kk = 0; kk < D_ENT; kk += 4) {
            v2f a, bb;
            a.x  = aRow[kk + koff]     * amask;
            a.y  = aRow[kk + koff + 1] * amask;
            bb.x = bRow[kk + koff];
            bb.y = bRow[kk + koff + 1];
            c = __builtin_amdgcn_wmma_f32_16x16x4_f32(
                    /*neg_a=*/false, a, /*neg_b=*/false, bb,
                    /*c_mod=*/(short)0, c, /*reuse_a=*/false, /*reuse_b=*/false);
        }
#pragma unroll
        for (int v = 0; v < 8; ++v)
            cTile[v + 8 * half][n0 + lm] = c[v];
    }
    __syncthreads();

    // ---- phase 2: bias + embeddings + mask + LayerNorm, one row per wave ----
    const int r = tile * 16 + wave;
    const int b = r / KN_LEN;
    const int k = r % KN_LEN;
    const float vmask = (float)kval[r];
    const float* te = tt_emb  + (size_t)ktt[r]  * HID;
    const float* pe = pos_emb + (size_t)kpos[r] * HID;

    float x[HID / 32];
    float s = 0.f, sq = 0.f;
#pragma unroll
    for (int t = 0; t < HID / 32; ++t) {
        const int j = lane + t * 32;
        const float v = (cTile[wave][j] + ke_b[j] + te[j] + pe[j]) * vmask;
        x[t] = v; s += v; sq += v * v;
    }
#pragma unroll
    for (int o = 16; o > 0; o >>= 1) {
        s  += __shfl_xor(s,  o, 32);
        sq += __shfl_xor(sq, o, 32);
    }
    const float mean = s * (1.f / HID);
    const float var  = sq * (1.f / HID) - mean * mean;
    const float rstd = rsqrtf(var + LN_EPS);
    float* op = out + ((size_t)b * SEQ + (WORD_LEN + k)) * HID;
#pragma unroll
    for (int t = 0; t < HID / 32; ++t) {
        const int j = lane + t * 32;
        op[j] = gamma[j] * (x[t] - mean) * rstd + beta[j];
    }
}

// ---------------------------------------------------------------------------
extern "C" void kernel_launch(void* const* d_in, const int* in_sizes, int n_in,
                              void* d_out, int out_size, void* d_ws, size_t ws_size,
                              hipStream_t stream)
{
    const int*   input_ids      = (const int*)d_in[0];
    const int*   token_type_ids = (const int*)d_in[1];
    const float* word_emb       = (const float*)d_in[2];
    const float* pos_emb        = (const float*)d_in[3];
    const float* tt_emb         = (const float*)d_in[4];
    const float* entityVec      = (const float*)d_in[5];
    const float* ke_w           = (const float*)d_in[6];
    const float* ke_b           = (const float*)d_in[7];
    const float* w_gamma        = (const float*)d_in[8];
    const float* w_beta         = (const float*)d_in[9];
    const float* k_gamma        = (const float*)d_in[10];
    const float* k_beta         = (const float*)d_in[11];
    float*       out            = (float*)d_out;

    // workspace layout: compacted index arrays (256 KB total)
    int* wid  = (int*)d_ws;
    int* wtt  = wid  + B_SZ * WORD_LEN;
    int* wpos = wtt  + B_SZ * WORD_LEN;
    int* kid  = wpos + B_SZ * WORD_LEN;
    int* ktt  = kid  + B_SZ * KN_LEN;
    int* kpos = ktt  + B_SZ * KN_LEN;
    int* kval = kpos + B_SZ * KN_LEN;

    compact_kernel<<<B_SZ, SEQ, 0, stream>>>(
        input_ids, token_type_ids, wid, wtt, wpos, kid, ktt, kpos, kval);

    word_kernel<<<B_SZ * WORD_LEN / 8, 256, 0, stream>>>(
        word_emb, pos_emb, tt_emb, w_gamma, w_beta, wid, wtt, wpos, out);

    kn_kernel<<<B_SZ * KN_LEN / 16, 512, 0, stream>>>(
        entityVec, ke_w, ke_b, tt_emb, pos_emb, k_gamma, k_beta,
        kid, ktt, kpos, kval, out);
}